// MyModel_22608707846181
// MI455X (gfx1250) — compile-verified
//
#include <hip/hip_runtime.h>
#include <math.h>

#define NN 2048
#define DD 64
#define KK 16
#define UNITS 64
#define ALPHA 0.3f
#define APAD 66   // LDS row pitch in floats (64 + 2 pad -> conflict-free column reads)

typedef __attribute__((ext_vector_type(2))) float v2f;
typedef __attribute__((ext_vector_type(8))) float v8f;
typedef __attribute__((ext_vector_type(4))) int   v4i;

__device__ __forceinline__ float lrelu(float x) { return x > 0.0f ? x : ALPHA * x; }

// --- gfx1250 async global->LDS staging (guarded; falls back to ld+st) -------
__device__ __forceinline__ void stage_to_lds_b128(const float* gsrc, float* ldst) {
#if __has_builtin(__builtin_amdgcn_global_load_async_to_lds_b128)
    __builtin_amdgcn_global_load_async_to_lds_b128(
        (__attribute__((address_space(1))) v4i*)gsrc,
        (__attribute__((address_space(3))) v4i*)ldst, 0, 0);
#else
    *(float4*)ldst = *(const float4*)gsrc;
#endif
}

__device__ __forceinline__ void wait_async_all() {
#if __has_builtin(__builtin_amdgcn_global_load_async_to_lds_b128)
#if __has_builtin(__builtin_amdgcn_s_wait_asynccnt)
    __builtin_amdgcn_s_wait_asynccnt(0);
#else
    asm volatile("s_wait_asynccnt 0" ::: "memory");
#endif
#endif
}

// ---------------------------------------------------------------------------
// K1: rel_weight[i][j] = lrelu(dot16(rel_encoding[i][j][:], W_rel) + b_rel)
// 256 MB streaming read; one thread per (i,j), 64B contiguous per thread.
// ---------------------------------------------------------------------------
__global__ void k_relweight(const float* __restrict__ rel_enc,
                            const float* __restrict__ W_rel,
                            const float* __restrict__ b_rel,
                            float* __restrict__ rel_w) {
    size_t t = (size_t)blockIdx.x * 256 + threadIdx.x;  // 0 .. N*N-1
    const float4* p = (const float4*)(rel_enc + t * KK);
    float4 e0 = p[0], e1 = p[1], e2 = p[2], e3 = p[3];
    const float4* w = (const float4*)W_rel;
    float4 w0 = w[0], w1 = w[1], w2 = w[2], w3 = w[3];
    float s = e0.x * w0.x + e0.y * w0.y + e0.z * w0.z + e0.w * w0.w;
    s += e1.x * w1.x + e1.y * w1.y + e1.z * w1.z + e1.w * w1.w;
    s += e2.x * w2.x + e2.y * w2.y + e2.z * w2.z + e2.w * w2.w;
    s += e3.x * w3.x + e3.y * w3.y + e3.z * w3.z + e3.w * w3.w;
    s += b_rel[0];
    rel_w[t] = lrelu(s);
}

// ---------------------------------------------------------------------------
// K2: logits[i][j] = rel_mask[i][j] + (F[i]·F[j]) * rel_weight[i][j]
// WMMA f32 16x16x4, fragments served from LDS staged via async-to-LDS.
// Block = 256 threads (8 waves): 8 row tiles (128 rows of A) x 1 col tile
// (16 rows of B, shared by all waves). grid = (N/16, N/128).
// ---------------------------------------------------------------------------
__global__ void k_gemm1(const float* __restrict__ F,
                        const float* __restrict__ rel_w,
                        const float* __restrict__ rel_mask,
                        float* __restrict__ wsW) {
    __shared__ float sA[128 * APAD];
    __shared__ float sB[16 * APAD];
    int tid = threadIdx.x;
    int tj = blockIdx.x;
    int rbase = blockIdx.y * 128;
    int j0 = tj * 16;

    // Stage A: 128 rows x 64 floats = 2048 float4-chunks, 8 per thread.
    for (int c = tid; c < 128 * 16; c += 256) {
        int row = c >> 4;
        int col = (c & 15) << 2;
        stage_to_lds_b128(F + (size_t)(rbase + row) * DD + col, &sA[row * APAD + col]);
    }
    // Stage B: 16 rows x 64 floats = 256 float4-chunks, 1 per thread.
    {
        int row = tid >> 4;
        int col = (tid & 15) << 2;
        stage_to_lds_b128(F + (size_t)(j0 + row) * DD + col, &sB[row * APAD + col]);
    }
    wait_async_all();
    __syncthreads();

    int lane = tid & 31, wave = tid >> 5;
    int lm = lane & 15, hi = lane >> 4;
    const float* aptr = &sA[(wave * 16 + lm) * APAD];  // A[m][k] = F[i0+m][k]
    const float* bptr = &sB[lm * APAD];                // B[k][n] = F[j0+n][k]
    v8f acc = {};
    for (int kb = 0; kb < DD; kb += 4) {
        v2f a = *(const v2f*)(aptr + kb + 2 * hi);
        v2f b = *(const v2f*)(bptr + kb + 2 * hi);
        acc = __builtin_amdgcn_wmma_f32_16x16x4_f32(false, a, false, b,
                                                    (short)0, acc, false, false);
    }
    // C/D layout: VGPR r, lanes 0-15 -> M=r, lanes 16-31 -> M=r+8; N=lm
    int i0 = rbase + wave * 16;
    for (int r = 0; r < 8; ++r) {
        int i = i0 + r + 8 * hi;
        size_t idx = (size_t)i * NN + j0 + lm;
        wsW[idx] = rel_mask[idx] + acc[r] * rel_w[idx];
    }
}

// ---------------------------------------------------------------------------
// K3: column softmax (axis=0): per column j, max over i, exp, store back,
// write invsum[j] = 1/sum. Thread-per-column gives coalesced row sweeps.
// grid = N/256, block = 256.
// ---------------------------------------------------------------------------
__global__ void k_softmax_cols(float* __restrict__ wsW,
                               float* __restrict__ invsum) {
    int j = blockIdx.x * 256 + threadIdx.x;
    float m = -3.4e38f;
    for (int i = 0; i < NN; ++i)
        m = fmaxf(m, wsW[(size_t)i * NN + j]);
    float s = 0.0f;
    for (int i = 0; i < NN; ++i) {
        float e = expf(wsW[(size_t)i * NN + j] - m);
        wsW[(size_t)i * NN + j] = e;
        s += e;
    }
    invsum[j] = 1.0f / s;
}

// ---------------------------------------------------------------------------
// K5: O = (E * invsum_k) @ F   where E = exp(logits), K-dim = 2048.
// Normalization folded into B operand: B'[k][n] = invsum[k] * F[k][j0+n].
// E is L2-resident (just written by K3), so direct loads are fine.
// grid = (D/16, N/16/8), block = 256 (8 waves).
// ---------------------------------------------------------------------------
__global__ void k_gemm2(const float* __restrict__ E,
                        const float* __restrict__ F,
                        const float* __restrict__ invs,
                        float* __restrict__ O) {
    int lane = threadIdx.x & 31;
    int wave = threadIdx.x >> 5;
    int tj = blockIdx.x;             // 0..3
    int ti = blockIdx.y * 8 + wave;  // 0..127
    int i0 = ti * 16, j0 = tj * 16;
    int lm = lane & 15, hi = lane >> 4;

    const float* arow = E + (size_t)(i0 + lm) * NN;
    v8f acc = {};
    for (int kb = 0; kb < NN; kb += 4) {
        int k = kb + 2 * hi;
        __builtin_prefetch(arow + kb + 64, 0, 0);  // global_prefetch_b8
        v2f a = *(const v2f*)(arow + k);
        v2f b;
        b.x = F[(size_t)k * DD + j0 + lm] * invs[k];
        b.y = F[(size_t)(k + 1) * DD + j0 + lm] * invs[k + 1];
        acc = __builtin_amdgcn_wmma_f32_16x16x4_f32(false, a, false, b,
                                                    (short)0, acc, false, false);
    }
    for (int r = 0; r < 8; ++r) {
        int i = i0 + r + 8 * hi;
        O[(size_t)i * DD + j0 + lm] = acc[r];
    }
}

// ---------------------------------------------------------------------------
// K6: per-row MLP. hidden[u] = lrelu(concat(F[i],O[i]) · W_h[:,u] + b_h[u]);
// pred[i] = lrelu(sum_u hidden[u]*W_p[u] + b_p). One 64-thread block per row.
// ---------------------------------------------------------------------------
__global__ void k_mlp(const float* __restrict__ F,
                      const float* __restrict__ O,
                      const float* __restrict__ W_h,
                      const float* __restrict__ b_h,
                      const float* __restrict__ W_p,
                      const float* __restrict__ b_p,
                      float* __restrict__ pred) {
    int i = blockIdx.x;
    int u = threadIdx.x;  // 0..63
    __shared__ float hs[UNITS];
    const float* f = F + (size_t)i * DD;
    const float* o = O + (size_t)i * DD;
    float s = b_h[u];
    for (int c = 0; c < DD; ++c) s += f[c] * W_h[c * UNITS + u];
    for (int c = 0; c < DD; ++c) s += o[c] * W_h[(DD + c) * UNITS + u];
    hs[u] = lrelu(s) * W_p[u];
    __syncthreads();
    for (int off = 32; off > 0; off >>= 1) {
        if (u < off) hs[u] += hs[u + off];
        __syncthreads();
    }
    if (u == 0) pred[i] = lrelu(hs[0] + b_p[0]);
}

// ---------------------------------------------------------------------------
extern "C" void kernel_launch(void* const* d_in, const int* in_sizes, int n_in,
                              void* d_out, int out_size, void* d_ws, size_t ws_size,
                              hipStream_t stream) {
    const float* Feature  = (const float*)d_in[0];
    const float* rel_enc  = (const float*)d_in[1];
    const float* rel_mask = (const float*)d_in[2];
    const float* W_rel    = (const float*)d_in[3];
    const float* b_rel    = (const float*)d_in[4];
    const float* W_h      = (const float*)d_in[5];
    const float* b_h      = (const float*)d_in[6];
    const float* W_p      = (const float*)d_in[7];
    const float* b_p      = (const float*)d_in[8];

    float* out   = (float*)d_out;
    float* rel_w = out;                    // [N,N]  (output 0)
    float* pred  = out + (size_t)NN * NN;  // [N]    (output 1)

    float* ws     = (float*)d_ws;
    float* wsW    = ws;                    // [N,N] logits -> exp values
    float* invsum = ws + (size_t)NN * NN;  // [N]
    float* wsO    = invsum + NN;           // [N,D] propagated outputs

    // K1: rel_weight straight into d_out
    k_relweight<<<(NN * NN) / 256, 256, 0, stream>>>(rel_enc, W_rel, b_rel, rel_w);

    // K2: logits via WMMA f32 GEMM (F@F^T) fused with mask + rel_weight scale
    dim3 g2(NN / 16, NN / 128);
    k_gemm1<<<g2, 256, 0, stream>>>(Feature, rel_w, rel_mask, wsW);

    // K3: column softmax (max, exp, 1/sum)
    k_softmax_cols<<<NN / 256, 256, 0, stream>>>(wsW, invsum);

    // K5: propagated features via WMMA f32 GEMM with fused normalization
    dim3 g5(DD / 16, NN / 16 / 8);
    k_gemm2<<<g5, 256, 0, stream>>>(wsW, Feature, invsum, wsO);

    // K6: final MLP -> prediction
    k_mlp<<<NN, UNITS, 0, stream>>>(Feature, wsO, W_h, b_h, W_p, b_p, pred);
}